// ConvFlow_37477884625310
// MI455X (gfx1250) — compile-verified
//
#include <hip/hip_runtime.h>
#include <math.h>

typedef __attribute__((ext_vector_type(2))) float v2f;
typedef __attribute__((ext_vector_type(8))) float v8f;

#define B_     16
#define C_IN   2
#define T_     8192
#define H_     192
#define KW     3
#define NB     10
#define CPOST  29
#define TILE   64
#define TBOUND 5.0f
#define MIN_Wf 0.001f
#define MIN_Hf 0.001f
#define MIN_Df 0.001f

// ---------------------------------------------------------------- pre: 1->192
__global__ __launch_bounds__(256) void pre_kernel(const float* __restrict__ x,
                                                  const float* __restrict__ pre_w,
                                                  const float* __restrict__ pre_b,
                                                  float* __restrict__ dst) {
  long idx = (long)blockIdx.x * 256 + threadIdx.x;   // over B*H*T
  int t = (int)(idx % T_);
  int o = (int)((idx / T_) % H_);
  int b = (int)(idx / ((long)T_ * H_));
  float xv = x[(long)b * C_IN * T_ + t];             // channel 0
  dst[idx] = pre_w[o] * xv + pre_b[o];
}

// ------------------------------------------------- per-column LN + exact GELU
__device__ __forceinline__ void ln_gelu(float* buf, const float* __restrict__ g,
                                        const float* __restrict__ bia,
                                        float* s_r1, float* s_r2,
                                        float* s_mu, float* s_rs, int tid) {
  const int col = tid & (TILE - 1);
  const int seg = tid >> 6;                    // 0..3
  float s = 0.f, s2 = 0.f;
  for (int i = 0; i < H_ / 4; ++i) {
    float v = buf[(seg * (H_ / 4) + i) * TILE + col];
    s += v; s2 += v * v;
  }
  s_r1[seg * TILE + col] = s;
  s_r2[seg * TILE + col] = s2;
  __syncthreads();
  if (tid < TILE) {
    float a = 0.f, a2 = 0.f;
    for (int k = 0; k < 4; ++k) { a += s_r1[k * TILE + tid]; a2 += s_r2[k * TILE + tid]; }
    float mu  = a * (1.0f / H_);
    float var = a2 * (1.0f / H_) - mu * mu;
    s_mu[tid] = mu;
    s_rs[tid] = rsqrtf(var + 1e-5f);
  }
  __syncthreads();
  for (int i = tid; i < H_ * TILE; i += 256) {
    int c = i / TILE, t = i % TILE;
    float xn = (buf[i] - s_mu[t]) * s_rs[t] * g[c] + bia[c];
    buf[i] = 0.5f * xn * (1.0f + erff(xn * 0.7071067811865476f));
  }
  __syncthreads();
}

// -------- fused layer: dwconv + LN/GELU + 192x192 GEMM (fp32 WMMA) + LN/GELU + res
__global__ __launch_bounds__(256) void layer_kernel(
    const float* __restrict__ src, float* __restrict__ dst,
    const float* __restrict__ dw,  const float* __restrict__ db,
    const float* __restrict__ g1,  const float* __restrict__ b1,
    const float* __restrict__ sw,  const float* __restrict__ sb,
    const float* __restrict__ g2,  const float* __restrict__ b2,
    int dil) {
  __shared__ float s_in[H_ * (TILE + 18)];   // 192*82 floats; reused as GEMM output
  __shared__ float s_mid[H_ * TILE];         // conv-out -> activated (GEMM B operand)
  __shared__ float s_r1[4 * TILE];
  __shared__ float s_r2[4 * TILE];
  __shared__ float s_mu[TILE];
  __shared__ float s_rs[TILE];

  const int tid = threadIdx.x;
  const int ntile = T_ / TILE;
  const int b  = blockIdx.x / ntile;
  const int t0 = (blockIdx.x % ntile) * TILE;
  const int pad = dil;
  const int W = TILE + 2 * pad;

  // 1) load tile + halo (zero padded)
  for (int i = tid; i < H_ * W; i += 256) {
    int c = i / W, j = i % W;
    int gt = t0 - pad + j;
    float v = 0.f;
    if (gt >= 0 && gt < T_) v = src[((long)b * H_ + c) * T_ + gt];
    s_in[c * W + j] = v;
  }
  __syncthreads();

  // 2) depthwise conv + bias
  for (int i = tid; i < H_ * TILE; i += 256) {
    int c = i / TILE, t = i % TILE;
    int ctr = t + pad;
    float v = dw[c * KW + 0] * s_in[c * W + ctr - dil]
            + dw[c * KW + 1] * s_in[c * W + ctr]
            + dw[c * KW + 2] * s_in[c * W + ctr + dil]
            + db[c];
    s_mid[i] = v;
  }
  __syncthreads();

  // 3) LN1 + GELU (in place on s_mid)
  ln_gelu(s_mid, g1, b1, s_r1, s_r2, s_mu, s_rs, tid);

  // 4) pointwise 192x192 GEMM via V_WMMA_F32_16X16X4_F32
  //    out[o,t] = sum_c sw[o,c] * act[c,t];  M=192 (12 tiles), N=64 (4 tiles), K=192
  //    Each wave task: one 16-row M tile x TWO adjacent 16-col N tiles sharing the
  //    same A fragment -> 2 v_wmma per global A load.
  float* s_out = s_in;               // halo buffer no longer needed
  const int lane  = tid & 31;
  const int wv    = tid >> 5;        // wave 0..7
  const int laneM = lane & 15;
  const int hi    = lane >> 4;       // K half select (A: K=2*hi,2*hi+1; rows M=laneM)
  for (int r = 0; r < 3; ++r) {
    int taskId = wv + 8 * r;         // 0..23
    int mt    = taskId >> 1;         // 0..11
    int npair = taskId & 1;          // 0 or 1
    int o0  = mt * 16;
    int tn0 = npair * 32;            // N tiles tn0 and tn0+16
    v8f acc0 = {0.f, 0.f, 0.f, 0.f, 0.f, 0.f, 0.f, 0.f};
    v8f acc1 = {0.f, 0.f, 0.f, 0.f, 0.f, 0.f, 0.f, 0.f};
    const float* arow  = sw + (long)(o0 + laneM) * H_ + 2 * hi;
    const float* brow0 = s_mid + tn0 + laneM;
    const float* brow1 = brow0 + 16;
#pragma unroll 4
    for (int k = 0; k < H_ / 4; ++k) {
      v2f a, bb0, bb1;
      a.x   = arow[k * 4 + 0];                   // A 16x4 f32 fragment (shared)
      a.y   = arow[k * 4 + 1];
      bb0.x = brow0[(k * 4 + 2 * hi + 0) * TILE];
      bb0.y = brow0[(k * 4 + 2 * hi + 1) * TILE];
      bb1.x = brow1[(k * 4 + 2 * hi + 0) * TILE];
      bb1.y = brow1[(k * 4 + 2 * hi + 1) * TILE];
      acc0 = __builtin_amdgcn_wmma_f32_16x16x4_f32(false, a, false, bb0,
                                                   (short)0, acc0, false, false);
      acc1 = __builtin_amdgcn_wmma_f32_16x16x4_f32(false, a, false, bb1,
                                                   (short)0, acc1, false, false);
    }
#pragma unroll
    for (int rr = 0; rr < 8; ++rr) {             // C/D layout: M = rr + 8*hi, N = laneM
      int M = o0 + rr + 8 * hi;
      float bias = sb[M];
      s_out[M * TILE + tn0 + laneM]      = acc0[rr] + bias;
      s_out[M * TILE + tn0 + 16 + laneM] = acc1[rr] + bias;
    }
  }
  __syncthreads();

  // 5) LN2 + GELU, then residual add and store
  ln_gelu(s_out, g2, b2, s_r1, s_r2, s_mu, s_rs, tid);
  for (int i = tid; i < H_ * TILE; i += 256) {
    int c = i / TILE, t = i % TILE;
    long gidx = ((long)b * H_ + c) * T_ + t0 + t;
    dst[gidx] = src[gidx] + s_out[i];
  }
}

// ------------------------------------- post matmul (192->29) + RQ spline + lad
__global__ __launch_bounds__(256) void post_spline_kernel(
    const float* __restrict__ hbuf, const float* __restrict__ x,
    const float* __restrict__ pw,   const float* __restrict__ pb,
    float* __restrict__ out, float* __restrict__ partial) {
  __shared__ float sred[256];
  const int tid = threadIdx.x;
  const int b = blockIdx.x >> 5;            // 32 tiles of 256 per batch
  const int t = (blockIdx.x & 31) * 256 + tid;

  float p[CPOST];
#pragma unroll
  for (int j = 0; j < CPOST; ++j) p[j] = pb[j];
  for (int c = 0; c < H_; ++c) {
    float hv = hbuf[((long)b * H_ + c) * T_ + t];
#pragma unroll
    for (int j = 0; j < CPOST; ++j) p[j] += pw[j * H_ + c] * hv;  // pw loads uniform -> SMEM
  }

  const float invsq = 0.07216878364870323f;   // 1/sqrt(192)
  float cw[NB + 1], wq[NB], ch[NB + 1], hq[NB], dd[NB + 1];

  { // widths
    float u[NB]; float mx = -1e30f;
#pragma unroll
    for (int j = 0; j < NB; ++j) { u[j] = p[j] * invsq; mx = fmaxf(mx, u[j]); }
    float s = 0.f;
#pragma unroll
    for (int j = 0; j < NB; ++j) { u[j] = expf(u[j] - mx); s += u[j]; }
    float inv = 1.0f / s, acc = 0.f;
    cw[0] = -TBOUND;
#pragma unroll
    for (int j = 0; j < NB; ++j) {
      acc += MIN_Wf + (1.0f - MIN_Wf * NB) * u[j] * inv;
      cw[j + 1] = -TBOUND + 2.0f * TBOUND * acc;
    }
    cw[NB] = TBOUND;
#pragma unroll
    for (int j = 0; j < NB; ++j) wq[j] = cw[j + 1] - cw[j];
  }
  { // heights
    float u[NB]; float mx = -1e30f;
#pragma unroll
    for (int j = 0; j < NB; ++j) { u[j] = p[NB + j] * invsq; mx = fmaxf(mx, u[j]); }
    float s = 0.f;
#pragma unroll
    for (int j = 0; j < NB; ++j) { u[j] = expf(u[j] - mx); s += u[j]; }
    float inv = 1.0f / s, acc = 0.f;
    ch[0] = -TBOUND;
#pragma unroll
    for (int j = 0; j < NB; ++j) {
      acc += MIN_Hf + (1.0f - MIN_Hf * NB) * u[j] * inv;
      ch[j + 1] = -TBOUND + 2.0f * TBOUND * acc;
    }
    ch[NB] = TBOUND;
#pragma unroll
    for (int j = 0; j < NB; ++j) hq[j] = ch[j + 1] - ch[j];
  }
  // derivatives: boundary pad makes d = MIN_D + softplus(log(expm1(1-MIN_D))) = 1.0
  dd[0] = 1.0f; dd[NB] = 1.0f;
#pragma unroll
  for (int j = 1; j < NB; ++j) {
    float uv = p[2 * NB + j - 1];
    float sp = (uv > 20.f) ? uv : log1pf(expf(uv));
    dd[j] = MIN_Df + sp;
  }

  const float x0v = x[((long)b * C_IN + 0) * T_ + t];
  const float xv  = x[((long)b * C_IN + 1) * T_ + t];
  const bool inside = (xv >= -TBOUND) && (xv <= TBOUND);
  const float xc = fminf(fmaxf(xv, -TBOUND), TBOUND);

  // fully unrolled bin select (no dynamic register indexing)
  float in_cw = cw[0], in_w = wq[0], in_ch = ch[0], in_h = hq[0];
  float d0 = dd[0], d1 = dd[1];
#pragma unroll
  for (int j = 1; j < NB; ++j) {
    bool sel = (xc >= cw[j]);
    in_cw = sel ? cw[j] : in_cw;  in_w = sel ? wq[j] : in_w;
    in_ch = sel ? ch[j] : in_ch;  in_h = sel ? hq[j] : in_h;
    d0    = sel ? dd[j] : d0;     d1   = sel ? dd[j + 1] : d1;
  }

  float delta = in_h / in_w;
  float theta = (xc - in_cw) / in_w;
  float t1m   = theta * (1.0f - theta);
  float denom = delta + (d0 + d1 - 2.0f * delta) * t1m;
  float num   = in_h * (delta * theta * theta + d0 * t1m);
  float y     = in_ch + num / denom;
  float omt   = 1.0f - theta;
  float dnum  = delta * delta * (d1 * theta * theta + 2.0f * delta * t1m + d0 * omt * omt);
  float lad   = logf(dnum) - 2.0f * logf(denom);

  out[((long)b * C_IN + 0) * T_ + t] = x0v;
  out[((long)b * C_IN + 1) * T_ + t] = inside ? y : xv;
  float ladv = inside ? lad : 0.0f;

  sred[tid] = ladv;
  __syncthreads();
  for (int s = 128; s > 0; s >>= 1) {
    if (tid < s) sred[tid] += sred[tid + s];
    __syncthreads();
  }
  if (tid == 0) partial[blockIdx.x] = sred[0];
}

// ---------------------------------------------- deterministic lad final reduce
__global__ void lad_reduce_kernel(const float* __restrict__ partial,
                                  float* __restrict__ lad_out) {
  int b = threadIdx.x;
  if (b < B_) {
    float s = 0.f;
    for (int j = 0; j < 32; ++j) s += partial[b * 32 + j];   // fixed order
    lad_out[b] = s;
  }
}

extern "C" void kernel_launch(void* const* d_in, const int* in_sizes, int n_in,
                              void* d_out, int out_size, void* d_ws, size_t ws_size,
                              hipStream_t stream) {
  const float* x       = (const float*)d_in[0];
  const float* pre_w   = (const float*)d_in[1];
  const float* pre_b   = (const float*)d_in[2];
  const float* dconv_w = (const float*)d_in[3];
  const float* dconv_b = (const float*)d_in[4];
  const float* ln1_g   = (const float*)d_in[5];
  const float* ln1_b   = (const float*)d_in[6];
  const float* sep_w   = (const float*)d_in[7];
  const float* sep_b   = (const float*)d_in[8];
  const float* ln2_g   = (const float*)d_in[9];
  const float* ln2_b   = (const float*)d_in[10];
  const float* post_w  = (const float*)d_in[11];
  const float* post_b  = (const float*)d_in[12];
  float* out = (float*)d_out;

  const long HBT = (long)B_ * H_ * T_;           // 25,165,824 floats = 96 MB
  float* buf0 = (float*)d_ws;
  float* buf1 = buf0 + HBT;
  float* partial = buf1 + HBT;

  pre_kernel<<<(B_ * H_ * T_) / 256, 256, 0, stream>>>(x, pre_w, pre_b, buf0);

  const float* srcs[3] = {buf0, buf1, buf0};
  float*       dsts[3] = {buf1, buf0, buf1};
  const int    dils[3] = {1, 3, 9};
  for (int i = 0; i < 3; ++i) {
    layer_kernel<<<B_ * (T_ / TILE), 256, 0, stream>>>(
        srcs[i], dsts[i],
        dconv_w + (long)i * H_ * KW, dconv_b + (long)i * H_,
        ln1_g + (long)i * H_, ln1_b + (long)i * H_,
        sep_w + (long)i * H_ * H_, sep_b + (long)i * H_,
        ln2_g + (long)i * H_, ln2_b + (long)i * H_, dils[i]);
  }

  post_spline_kernel<<<B_ * 32, 256, 0, stream>>>(buf1, x, post_w, post_b, out, partial);
  lad_reduce_kernel<<<1, 32, 0, stream>>>(partial, out + (long)B_ * C_IN * T_);
}